// CombinedDynamicMarginLoss_arc_4526895529994
// MI455X (gfx1250) — compile-verified
//
#include <hip/hip_runtime.h>
#include <hip/hip_bf16.h>
#include <math.h>
#include <stdint.h>

// CombinedDynamicMarginLoss_arc forward, MI455X (gfx1250).
// Memory-bound streaming kernel: 1.64 GB total traffic -> ~70us floor @ 23.3 TB/s.
// Uses CDNA5 async global->LDS b128 loads (ASYNCcnt) for a 4-deep prefetch
// pipeline and non-temporal b128 stores for the write stream.

typedef float v4f __attribute__((ext_vector_type(4)));

#define THREADS 256
#define DEPTH   4          // async pipeline depth (LDS staging buffers)

#define SCALE_S 64.0f
#define MARGIN2 0.5f
#define ALPHA_F 0.1f
#define NEG_BIG (-1.0e9f)

__device__ __forceinline__ void async_load_b128(uint32_t lds_off, uint64_t gaddr) {
  // CDNA5 async tensor-path load: 16B per lane, global memory -> LDS,
  // tracked with ASYNCcnt (no VGPR data movement).
  asm volatile("global_load_async_to_lds_b128 %0, %1, off"
               :: "v"(lds_off), "v"(gaddr)
               : "memory");
}

__device__ __forceinline__ void wait_asynccnt(int n) {
  // SIMM16 must be a literal; small switch covers pipeline drain states.
  switch (n) {
    case 0:  asm volatile("s_wait_asynccnt 0x0" ::: "memory"); break;
    case 1:  asm volatile("s_wait_asynccnt 0x1" ::: "memory"); break;
    case 2:  asm volatile("s_wait_asynccnt 0x2" ::: "memory"); break;
    default: asm volatile("s_wait_asynccnt 0x3" ::: "memory"); break;
  }
}

__global__ __launch_bounds__(THREADS)
void arc_margin_kernel(const float* __restrict__ logits,
                       const int* __restrict__ labels,
                       float* __restrict__ out,
                       int C) {
  const int row = blockIdx.x;
  const int tid = threadIdx.x;
  const int C4  = C >> 2;                       // C is a multiple of 4 (50000)
  const int T   = (C4 + THREADS - 1) / THREADS; // tiles per row

  __shared__ v4f   stage[DEPTH][THREADS];       // 16 KB staging (16B aligned)
  __shared__ float smax[THREADS];
  __shared__ float s_cosy;

  const int  lraw  = labels[row];
  const bool valid = (lraw != -1);
  const int  lab   = valid ? lraw : -2;         // -2 never matches a column

  const float* __restrict__ src = logits + (size_t)row * (size_t)C;
  float*       __restrict__ dst = out    + (size_t)row * (size_t)C;

  // ---- Prologue: fill the async pipeline -------------------------------
  int issued = 0;
  for (; issued < DEPTH && issued < T; ++issued) {
    int j4  = issued * THREADS + tid;
    int j4c = (j4 < C4) ? j4 : (C4 - 1);        // clamp tail (uniform EXEC)
    async_load_b128((uint32_t)(uintptr_t)&stage[issued % DEPTH][tid],
                    (uint64_t)(uintptr_t)(src + 4 * (size_t)j4c));
  }

  // ---- Main stream loop -------------------------------------------------
  float mx = NEG_BIG;
  for (int t = 0; t < T; ++t) {
    if (issued < T) {
      int j4  = issued * THREADS + tid;
      int j4c = (j4 < C4) ? j4 : (C4 - 1);
      async_load_b128((uint32_t)(uintptr_t)&stage[issued % DEPTH][tid],
                      (uint64_t)(uintptr_t)(src + 4 * (size_t)j4c));
      ++issued;
    }
    // tiles t..issued-1 are in flight; need tile t complete
    wait_asynccnt(issued - t - 1);

    int j4 = t * THREADS + tid;
    if (j4 < C4) {
      v4f v = stage[t % DEPTH][tid];            // each lane reads only its own slot
      int base = 4 * j4;
      bool hasLab = (lab >= base) & (lab < base + 4);

#pragma unroll
      for (int k = 0; k < 4; ++k) {
        float e = v[k];
        mx = fmaxf(mx, (base + k == lab) ? NEG_BIG : e);
      }

      if (!hasLab) {
        v4f o = v * SCALE_S;
        __builtin_nontemporal_store(o, (v4f*)(dst + base)); // streamed, no reuse
      } else {
        // label column is written exactly once, later, by thread 0
#pragma unroll
        for (int k = 0; k < 4; ++k) {
          if (base + k == lab) s_cosy = v[k];
          else __builtin_nontemporal_store(v[k] * SCALE_S, dst + base + k);
        }
      }
    }
  }

  // ---- Block max reduction ---------------------------------------------
  smax[tid] = mx;
  __syncthreads();
#pragma unroll
  for (int s = THREADS / 2; s > 0; s >>= 1) {
    if (tid < s) smax[tid] = fmaxf(smax[tid], smax[tid + s]);
    __syncthreads();
  }

  // ---- Per-row margin fixup (tiny trig; thread 0 only) ------------------
  if (tid == 0 && valid) {
    const float PI_F = 3.14159265358979323846f;
    float cy = s_cosy;
    float mo = smax[0];
    float ty = acosf(fminf(fmaxf(cy, -1.0f), 1.0f));
    float tm = acosf(fminf(fmaxf(mo, -1.0f), 1.0f));
    float h  = 0.5f * PI_F - (tm - ty);
    h = fminf(fmaxf(h, 0.0f), PI_F / 3.0f);
    float mi  = MARGIN2 + ALPHA_F * h;          // m2 + alpha*h
    float phi = cosf(ty + mi);                  // m1 = 1, m3 = 0
    float fin = fminf(phi, cy);
    dst[lab] = fin * SCALE_S;
  }
}

extern "C" void kernel_launch(void* const* d_in, const int* in_sizes, int n_in,
                              void* d_out, int out_size, void* d_ws, size_t ws_size,
                              hipStream_t stream) {
  const float* logits = (const float*)d_in[0];
  const int*   labels = (const int*)d_in[1];
  float*       out    = (float*)d_out;

  const int N = in_sizes[1];                 // 4096 rows
  const int C = in_sizes[0] / N;             // 50000 classes (multiple of 4)

  arc_margin_kernel<<<N, THREADS, 0, stream>>>(logits, labels, out, C);
}